// DCNNv2_17111149707558
// MI455X (gfx1250) — compile-verified
//
#include <hip/hip_runtime.h>

typedef __attribute__((ext_vector_type(2))) float v2f;
typedef __attribute__((ext_vector_type(8))) float v8f;
typedef int v4i __attribute__((vector_size(16)));
typedef __attribute__((address_space(1))) v4i* gptr_v4i;
typedef __attribute__((address_space(3))) v4i* lptr_v4i;

namespace {
constexpr int kG = 32768;
constexpr int kN = 128;
constexpr int kE = 1024;
constexpr int kK = 32;
constexpr int kD = 3;
constexpr int kB = kG / 2;
constexpr int kThreads = 256;
}

// One block (8 wave32) per graph: segment-sum in LDS, then per-wave
// V_WMMA_F32_16X16X4_F32 tiles for relu(x@W + agg@M) column sums,
// softmax chain, writes per-graph 3-vector embedding to d_ws.
__global__ __launch_bounds__(kThreads) void dcnn_graph_kernel(
    const float* __restrict__ x,
    const int*   __restrict__ edge_src,
    const int*   __restrict__ edge_dst,
    const float* __restrict__ ext_x,
    const float* __restrict__ Wp,
    const float* __restrict__ Mp,
    const float* __restrict__ Up,
    const float* __restrict__ Vp,
    float* __restrict__ gemb)   // [G, 3]
{
  __shared__ __align__(16) float x_lds[kN * kD];    // 384 floats
  __shared__ float agg_lds[kN * kD];                // 384 floats
  __shared__ float wm_lds[2 * kD * kD];             // W (9) then M (9)
  __shared__ float hsum[4];
  __shared__ float extsum[4];

  const int g   = blockIdx.x;
  const int tid = threadIdx.x;
  const float* __restrict__ xg = x + (size_t)g * (kN * kD);

  // ---- stage x tile into LDS (async-to-LDS path when available) ----
#if __has_builtin(__builtin_amdgcn_global_load_async_to_lds_b128)
  if (tid < (kN * kD) / 4) {  // 96 lanes == 3 full waves, 16B each
    __builtin_amdgcn_global_load_async_to_lds_b128(
        (gptr_v4i)(xg + tid * 4),
        (lptr_v4i)(&x_lds[tid * 4]),
        0, 0);
  }
#if __has_builtin(__builtin_amdgcn_s_wait_asynccnt)
  __builtin_amdgcn_s_wait_asynccnt(0);
#else
  asm volatile("s_wait_asynccnt 0x0" ::: "memory");
#endif
#else
  for (int i = tid; i < kN * kD; i += kThreads) x_lds[i] = xg[i];
#endif

  for (int i = tid; i < kN * kD; i += kThreads) agg_lds[i] = 0.0f;
  if (tid < 2 * kD * kD) wm_lds[tid] = (tid < kD * kD) ? Wp[tid] : Mp[tid - kD * kD];
  if (tid < kD) { hsum[tid] = 0.0f; extsum[tid] = 0.0f; }
  __syncthreads();

  // ---- segment sum over edges, entirely in LDS (ds_add_f32) ----
  const int* __restrict__ esrc = edge_src + (size_t)g * kE;
  const int* __restrict__ edst = edge_dst + (size_t)g * kE;
  for (int e = tid; e < kE; e += kThreads) {
    if (e + kThreads < kE) {
      __builtin_prefetch(&esrc[e + kThreads], 0, 3);
      __builtin_prefetch(&edst[e + kThreads], 0, 3);
    }
    const int s = esrc[e];
    const int d = edst[e];
    const float v0 = x_lds[s * 3 + 0];
    const float v1 = x_lds[s * 3 + 1];
    const float v2 = x_lds[s * 3 + 2];
    atomicAdd(&agg_lds[d * 3 + 0], v0);
    atomicAdd(&agg_lds[d * 3 + 1], v1);
    atomicAdd(&agg_lds[d * 3 + 2], v2);
  }
  __syncthreads();

  // ---- per-wave WMMA: D = x@W + agg@M over this wave's 16 nodes ----
  // A layout (32-bit 16x4): lanes 0-15 hold M=0..15 / K={0,1};
  //                         lanes 16-31 hold M=0..15 / K={2,3}.
  // B layout (32-bit 4x16): VGPR v, lane group g16 -> K = 2*g16 + v, N = lane%16.
  const int lane = tid & 31;
  const int half = lane >> 4;
  const int l16  = lane & 15;
  const int wave = tid >> 5;
  const int row  = wave * 16 + l16;   // node index 0..127

  v2f ax, aagg, bw, bm;
  ax.x   = (half == 0) ? x_lds[row * 3 + 0]   : x_lds[row * 3 + 2];
  ax.y   = (half == 0) ? x_lds[row * 3 + 1]   : 0.0f;   // K=3 pad
  aagg.x = (half == 0) ? agg_lds[row * 3 + 0] : agg_lds[row * 3 + 2];
  aagg.y = (half == 0) ? agg_lds[row * 3 + 1] : 0.0f;

  float bwx = 0.0f, bwy = 0.0f, bmx = 0.0f, bmy = 0.0f;
  if (l16 < kD) {
    if (half == 0) {
      bwx = wm_lds[0 * 3 + l16]; bwy = wm_lds[1 * 3 + l16];       // W rows K=0,1
      bmx = wm_lds[9 + 0 * 3 + l16]; bmy = wm_lds[9 + 1 * 3 + l16];
    } else {
      bwx = wm_lds[2 * 3 + l16];                                   // W row K=2, K=3 pad
      bmx = wm_lds[9 + 2 * 3 + l16];
    }
  }
  bw.x = bwx; bw.y = bwy; bm.x = bmx; bm.y = bmy;

  v8f c = {};
  c = __builtin_amdgcn_wmma_f32_16x16x4_f32(false, ax,   false, bw, (short)0, c, false, false);
  c = __builtin_amdgcn_wmma_f32_16x16x4_f32(false, aagg, false, bm, (short)0, c, false, false);

  // relu + column-sum: lane holds column N=l16, rows (half?8:0)+j
  float colsum = 0.0f;
#pragma unroll
  for (int j = 0; j < 8; ++j) colsum += fmaxf(c[j], 0.0f);
  if (l16 < kD) atomicAdd(&hsum[l16], colsum);

  // ---- ext_x column sums (96 floats) ----
  const float* __restrict__ exg = ext_x + (size_t)g * (kK * kD);
  for (int t = tid; t < kK * kD; t += kThreads) atomicAdd(&extsum[t % kD], exg[t]);
  __syncthreads();

  // ---- per-graph scalar tail: softmax -> relu(U,V mix) -> softmax ----
  if (tid == 0) {
    const float h0 = hsum[0], h1 = hsum[1], h2 = hsum[2];
    const float mh = fmaxf(h0, fmaxf(h1, h2));
    const float e0 = __expf(h0 - mh), e1 = __expf(h1 - mh), e2 = __expf(h2 - mh);
    const float ei = 1.0f / (e0 + e1 + e2);
    const float emb0 = e0 * ei, emb1 = e1 * ei, emb2 = e2 * ei;
    const float s0 = extsum[0], s1 = extsum[1], s2 = extsum[2];
    float ex[3];
#pragma unroll
    for (int d = 0; d < 3; ++d) {
      const float v = emb0 * Up[0 * 3 + d] + emb1 * Up[1 * 3 + d] + emb2 * Up[2 * 3 + d]
                    + s0 * Vp[0 * 3 + d] + s1 * Vp[1 * 3 + d] + s2 * Vp[2 * 3 + d];
      ex[d] = fmaxf(v, 0.0f);
    }
    const float mg = fmaxf(ex[0], fmaxf(ex[1], ex[2]));
    const float g0 = __expf(ex[0] - mg), g1 = __expf(ex[1] - mg), g2 = __expf(ex[2] - mg);
    const float gi = 1.0f / (g0 + g1 + g2);
    gemb[(size_t)g * 3 + 0] = g0 * gi;
    gemb[(size_t)g * 3 + 1] = g1 * gi;
    gemb[(size_t)g * 3 + 2] = g2 * gi;
  }
}

// Pairing MLP: t = [e1*e2, e1+e2] (6) -> relu(W1+b1) (3) -> W2+b2 (2) -> softmax
__global__ __launch_bounds__(kThreads) void dcnn_pair_kernel(
    const float* __restrict__ gemb,
    const float* __restrict__ W1, const float* __restrict__ b1,
    const float* __restrict__ W2, const float* __restrict__ b2,
    float* __restrict__ out)
{
  const int i = blockIdx.x * kThreads + threadIdx.x;
  if (i >= kB) return;
  const float* __restrict__ p1 = gemb + (size_t)i * 3;
  const float* __restrict__ p2 = gemb + (size_t)(i + kB) * 3;
  const float a0 = p1[0], a1 = p1[1], a2 = p1[2];
  const float c0 = p2[0], c1 = p2[1], c2 = p2[2];
  const float t[6] = {a0 * c0, a1 * c1, a2 * c2, a0 + c0, a1 + c1, a2 + c2};

  float h[3];
#pragma unroll
  for (int d = 0; d < 3; ++d) {
    float s = b1[d];
#pragma unroll
    for (int j = 0; j < 6; ++j) s += t[j] * W1[j * 3 + d];
    h[d] = fmaxf(s, 0.0f);
  }
  const float o0 = b2[0] + h[0] * W2[0] + h[1] * W2[2] + h[2] * W2[4];
  const float o1 = b2[1] + h[0] * W2[1] + h[1] * W2[3] + h[2] * W2[5];
  const float m = fmaxf(o0, o1);
  const float e0 = __expf(o0 - m), e1 = __expf(o1 - m);
  const float inv = 1.0f / (e0 + e1);
  out[(size_t)i * 2 + 0] = e0 * inv;
  out[(size_t)i * 2 + 1] = e1 * inv;
}

extern "C" void kernel_launch(void* const* d_in, const int* in_sizes, int n_in,
                              void* d_out, int out_size, void* d_ws, size_t ws_size,
                              hipStream_t stream) {
  const float* x        = (const float*)d_in[0];
  const int*   edge_src = (const int*)d_in[1];
  const int*   edge_dst = (const int*)d_in[2];
  const float* ext_x    = (const float*)d_in[3];
  const float* W        = (const float*)d_in[4];
  const float* M        = (const float*)d_in[5];
  const float* U        = (const float*)d_in[6];
  const float* V        = (const float*)d_in[7];
  const float* W1       = (const float*)d_in[8];
  const float* b1       = (const float*)d_in[9];
  const float* W2       = (const float*)d_in[10];
  const float* b2       = (const float*)d_in[11];
  float* out  = (float*)d_out;
  float* gemb = (float*)d_ws;   // needs kG*3*4 = 384 KiB of scratch

  dcnn_graph_kernel<<<kG, kThreads, 0, stream>>>(
      x, edge_src, edge_dst, ext_x, W, M, U, V, gemb);
  dcnn_pair_kernel<<<kB / kThreads, kThreads, 0, stream>>>(
      gemb, W1, b1, W2, b2, out);
}